// DCRNN_39788577030952
// MI455X (gfx1250) — compile-verified
//
#include <hip/hip_runtime.h>
#include <hip/hip_bf16.h>
#include <stdint.h>

// ---------------------------------------------------------------------------
// DCRNN forward for gfx1250 (MI455X).
// Dense GEMMs: V_WMMA_F32_16X16X4_F32, one 16x16 tile per wave32, K=144 fused
//   [XH | T1o | T1i] operand, weights staged in LDS via async global->LDS copy.
// Sparse diffusion: float4-vectorized gather + global atomic adds (BW-bound).
// ---------------------------------------------------------------------------

typedef __attribute__((ext_vector_type(2))) float v2f;
typedef __attribute__((ext_vector_type(8))) float v8f;

#define N_F   16
#define N_P   6
#define N_O   32
#define N_C   48   // F + O
#define KTOT  144  // 3 * C

#define WIDX(g,d,k,c,o) (((((g)*2+(d))*2+(k))*48+(c))*32+(o))

// --------------------- async global -> LDS staging helpers ------------------
// For LDS-aperture generic addresses, addr[31:0] is the LDS byte offset
// (CDNA5 ISA 10.2), so the low 32 bits of a __shared__ pointer are the VDST
// operand of global_load_async_to_lds_*.  Tracked by ASYNCcnt.
__device__ __forceinline__ void async_copy_b128(const float* g, float* l) {
  unsigned lds = (unsigned)(uintptr_t)l;
  unsigned long long ga = (unsigned long long)(uintptr_t)g;
  asm volatile("global_load_async_to_lds_b128 %0, %1, off"
               :: "v"(lds), "v"(ga) : "memory");
}
__device__ __forceinline__ void async_wait0() {
  asm volatile("s_wait_asynccnt 0x0" ::: "memory");
}

// ------------------------------- utility -----------------------------------

__global__ void zero_kernel(float* __restrict__ p, int n) {
  int i = blockIdx.x * blockDim.x + threadIdx.x;
  if (i < n) p[i] = 0.0f;
}

// weighted degrees
__global__ void deg_kernel(const int* __restrict__ src, const int* __restrict__ dst,
                           const float* __restrict__ w,
                           float* __restrict__ dego, float* __restrict__ degi, int E) {
  int e = blockIdx.x * blockDim.x + threadIdx.x;
  if (e < E) {
    float we = w[e];
    atomicAdd(&dego[src[e]], we);
    atomicAdd(&degi[dst[e]], we);
  }
}

// random-walk normalized edge weights (safe inverse)
__global__ void norm_kernel(const int* __restrict__ src, const int* __restrict__ dst,
                            const float* __restrict__ w,
                            const float* __restrict__ dego, const float* __restrict__ degi,
                            float* __restrict__ wno, float* __restrict__ wni, int E) {
  int e = blockIdx.x * blockDim.x + threadIdx.x;
  if (e < E) {
    float we = w[e];
    float d0 = dego[src[e]];
    float d1 = degi[dst[e]];
    wno[e] = (d0 > 0.0f) ? we / d0 : 0.0f;
    wni[e] = (d1 > 0.0f) ? we / d1 : 0.0f;
  }
}

// ------------------------------ weight packing ------------------------------
// W: [3,2,K=2,48,32].  Packed GEMM operand rows: 0..47 -> W[g,0,0]+W[g,1,0],
// 48..95 -> W[g,0,1], 96..143 -> W[g,1,1].
// ZR pack: [144,64], cols 0..31 = gate Z (g=0), 32..63 = gate R (g=1).
__global__ void pack_zr_kernel(const float* __restrict__ W, const float* __restrict__ b,
                               float* __restrict__ Wzr, float* __restrict__ bzr) {
  int idx = blockIdx.x * blockDim.x + threadIdx.x;
  if (idx < KTOT * 64) {
    int r = idx >> 6, col = idx & 63;
    int g = col >> 5, o = col & 31;
    float v;
    if (r < 48)       v = W[WIDX(g,0,0,r,o)] + W[WIDX(g,1,0,r,o)];
    else if (r < 96)  v = W[WIDX(g,0,1,r-48,o)];
    else              v = W[WIDX(g,1,1,r-96,o)];
    Wzr[idx] = v;
  }
  if (idx < 64) bzr[idx] = b[(idx >> 5) * 32 + (idx & 31)];
}

// H-candidate pack (g=2): [144,32]
__global__ void pack_h_kernel(const float* __restrict__ W, const float* __restrict__ b,
                              float* __restrict__ Wh, float* __restrict__ bh) {
  int idx = blockIdx.x * blockDim.x + threadIdx.x;
  if (idx < KTOT * 32) {
    int r = idx >> 5, o = idx & 31;
    float v;
    if (r < 48)       v = W[WIDX(2,0,0,r,o)] + W[WIDX(2,1,0,r,o)];
    else if (r < 96)  v = W[WIDX(2,0,1,r-48,o)];
    else              v = W[WIDX(2,1,1,r-96,o)];
    Wh[idx] = v;
  }
  if (idx < 32) bh[idx] = b[2 * 32 + idx];
}

// ------------------------------ concat builders -----------------------------

// XH[n,0:16] = x[n,f,t] (x is [N,F,P]); XH[n,16:48] = H[n,:]
__global__ void build_xh_x_kernel(const float* __restrict__ x, const float* __restrict__ H,
                                  float* __restrict__ XH, int t, int N) {
  int idx = blockIdx.x * blockDim.x + threadIdx.x;
  if (idx >= N * N_C) return;
  int n = idx / N_C, c = idx % N_C;
  XH[idx] = (c < N_F) ? x[(size_t)n * (N_F * N_P) + c * N_P + t]
                      : H[(size_t)n * N_O + (c - N_F)];
}

// XH[n,0:16] = outb[n,:]; XH[n,16:48] = H[n,:]
__global__ void build_xh_out_kernel(const float* __restrict__ outb, const float* __restrict__ H,
                                    float* __restrict__ XH, int N) {
  int idx = blockIdx.x * blockDim.x + threadIdx.x;
  if (idx >= N * N_C) return;
  int n = idx / N_C, c = idx % N_C;
  XH[idx] = (c < N_F) ? outb[(size_t)n * N_F + c]
                      : H[(size_t)n * N_O + (c - N_F)];
}

// overwrite hidden part of XH with R*H (X part unchanged)
__global__ void update_rh_kernel(const float* __restrict__ R, const float* __restrict__ H,
                                 float* __restrict__ XH, int N) {
  int idx = blockIdx.x * blockDim.x + threadIdx.x;
  if (idx >= N * N_O) return;
  int n = idx / N_O, o = idx % N_O;
  XH[(size_t)n * N_C + N_F + o] = R[idx] * H[idx];
}

// decoder initial output: out = x[:,:,P-1]
__global__ void init_out_kernel(const float* __restrict__ x, float* __restrict__ outb, int N) {
  int idx = blockIdx.x * blockDim.x + threadIdx.x;
  if (idx >= N * N_F) return;
  int n = idx / N_F, f = idx % N_F;
  outb[idx] = x[(size_t)n * (N_F * N_P) + f * N_P + (N_P - 1)];
}

// ------------------------------ sparse diffusion ----------------------------
// To[dst[e]] += wno[e]*X[src[e]] ; Ti[src[e]] += wni[e]*X[dst[e]]
// one thread per (edge, float4 chunk of 48 features)
__global__ void prop_kernel(const float* __restrict__ X,
                            const int* __restrict__ src, const int* __restrict__ dst,
                            const float* __restrict__ wno, const float* __restrict__ wni,
                            float* __restrict__ To, float* __restrict__ Ti, int E) {
  int idx = blockIdx.x * blockDim.x + threadIdx.x;
  int e = idx / 12;
  if (e >= E) return;
  int c = (idx % 12) * 4;
  int s = src[e], d = dst[e];
  float wo = wno[e], wi = wni[e];
  const float4 xs = *(const float4*)(X + (size_t)s * N_C + c);
  const float4 xd = *(const float4*)(X + (size_t)d * N_C + c);
  float* to = To + (size_t)d * N_C + c;
  float* ti = Ti + (size_t)s * N_C + c;
  atomicAdd(to + 0, wo * xs.x);
  atomicAdd(to + 1, wo * xs.y);
  atomicAdd(to + 2, wo * xs.z);
  atomicAdd(to + 3, wo * xs.w);
  atomicAdd(ti + 0, wi * xd.x);
  atomicAdd(ti + 1, wi * xd.y);
  atomicAdd(ti + 2, wi * xd.z);
  atomicAdd(ti + 3, wi * xd.w);
}

// ------------------------------ WMMA GEMM cores -----------------------------
// A fragment (16x4 f32): lanes 0-15 hold K={k,k+1}, lanes 16-31 hold K={k+2,k+3},
// row M = lane%16 (CDNA5 ISA 7.12.2). B mirrors with N = lane%16.
// Virtual A = [XH | T1o | T1i] (K = 144); the three 48-wide segments are
// iterated branch-free, A fragments preloaded per segment to batch the loads.

// Z/R gate GEMM: out width 64 (cols 0-31 -> Z, 32-63 -> R), sigmoid applied.
// 8 waves: 4 N-tiles x 2 M-tiles -> 32 rows/block.
__global__ void gemm_zr_kernel(const float* __restrict__ XH, const float* __restrict__ T1o,
                               const float* __restrict__ T1i,
                               const float* __restrict__ Wzr, const float* __restrict__ bzr,
                               float* __restrict__ Z, float* __restrict__ R, int N) {
  __shared__ __align__(16) float sW[KTOT * 64];
  __shared__ float sB[64];
  for (int i = threadIdx.x * 4; i < KTOT * 64; i += blockDim.x * 4)
    async_copy_b128(Wzr + i, sW + i);
  if (threadIdx.x < 64) sB[threadIdx.x] = bzr[threadIdx.x];
  async_wait0();
  __syncthreads();

  const int wave = threadIdx.x >> 5, lane = threadIdx.x & 31;
  const int half = lane >> 4, lm = lane & 15;
  const int ntile = wave & 3, mtile = wave >> 2;
  const int row0 = blockIdx.x * 32 + mtile * 16;
  const int n0 = ntile * 16;
  const int arow = row0 + lm;

  const float* const segs[3] = { XH, T1o, T1i };
  v8f acc = {};
#pragma unroll
  for (int seg = 0; seg < 3; ++seg) {
    const float* ap = segs[seg] + (size_t)arow * N_C + half * 2;
    v2f a[12];
#pragma unroll
    for (int t = 0; t < 12; ++t) a[t] = *(const v2f*)(ap + 4 * t);
#pragma unroll
    for (int t = 0; t < 12; ++t) {
      const int k = seg * 48 + 4 * t + half * 2;
      v2f b;
      b.x = sW[(k + 0) * 64 + n0 + lm];
      b.y = sW[(k + 1) * 64 + n0 + lm];
      acc = __builtin_amdgcn_wmma_f32_16x16x4_f32(false, a[t], false, b, (short)0,
                                                  acc, false, false);
    }
  }
#pragma unroll
  for (int r = 0; r < 8; ++r) {
    int grow = row0 + r + 8 * half;
    int col = n0 + lm;
    float v = acc[r] + sB[col];
    v = 1.0f / (1.0f + __expf(-v));
    if (col < 32) Z[(size_t)grow * N_O + col] = v;
    else          R[(size_t)grow * N_O + (col - 32)] = v;
  }
}

// H-candidate GEMM + GRU merge: Ht = tanh(.), Hnew = Z*H + (1-Z)*Ht (+ReLU enc).
// 8 waves: 2 N-tiles x 4 M-tiles -> 64 rows/block.
__global__ void gemm_h_kernel(const float* __restrict__ XR, const float* __restrict__ T1o,
                              const float* __restrict__ T1i,
                              const float* __restrict__ Wh, const float* __restrict__ bh,
                              const float* __restrict__ Z, float* __restrict__ H,
                              int relu, int N) {
  __shared__ __align__(16) float sW[KTOT * 32];
  __shared__ float sB[32];
  for (int i = threadIdx.x * 4; i < KTOT * 32; i += blockDim.x * 4)
    async_copy_b128(Wh + i, sW + i);
  if (threadIdx.x < 32) sB[threadIdx.x] = bh[threadIdx.x];
  async_wait0();
  __syncthreads();

  const int wave = threadIdx.x >> 5, lane = threadIdx.x & 31;
  const int half = lane >> 4, lm = lane & 15;
  const int ntile = wave & 1, mtile = wave >> 1;
  const int row0 = blockIdx.x * 64 + mtile * 16;
  const int n0 = ntile * 16;
  const int arow = row0 + lm;

  const float* const segs[3] = { XR, T1o, T1i };
  v8f acc = {};
#pragma unroll
  for (int seg = 0; seg < 3; ++seg) {
    const float* ap = segs[seg] + (size_t)arow * N_C + half * 2;
    v2f a[12];
#pragma unroll
    for (int t = 0; t < 12; ++t) a[t] = *(const v2f*)(ap + 4 * t);
#pragma unroll
    for (int t = 0; t < 12; ++t) {
      const int k = seg * 48 + 4 * t + half * 2;
      v2f b;
      b.x = sW[(k + 0) * 32 + n0 + lm];
      b.y = sW[(k + 1) * 32 + n0 + lm];
      acc = __builtin_amdgcn_wmma_f32_16x16x4_f32(false, a[t], false, b, (short)0,
                                                  acc, false, false);
    }
  }
#pragma unroll
  for (int r = 0; r < 8; ++r) {
    int grow = row0 + r + 8 * half;
    int col = n0 + lm;
    float ht = tanhf(acc[r] + sB[col]);
    size_t p = (size_t)grow * N_O + col;
    float z = Z[p];
    float hv = z * H[p] + (1.0f - z) * ht;
    if (relu) hv = fmaxf(hv, 0.0f);
    H[p] = hv;
  }
}

// Decoder output GEMM: out += H @ lin_W^T + lin_b, also scatter into d_out[:, :, t].
// B[o][f] = lin_W[f,o]. 8 waves = 8 M-tiles -> 128 rows/block (tail-guarded).
__global__ void gemm_lin_kernel(const float* __restrict__ H, const float* __restrict__ linW,
                                const float* __restrict__ linb,
                                float* __restrict__ outb, float* __restrict__ dout,
                                int t, int N) {
  __shared__ float sBm[N_O * N_F];
  __shared__ float sB[N_F];
  for (int i = threadIdx.x; i < N_O * N_F; i += blockDim.x) {
    int o = i >> 4, f = i & 15;
    sBm[i] = linW[f * N_O + o];   // transpose: B[o][f]
  }
  if (threadIdx.x < N_F) sB[threadIdx.x] = linb[threadIdx.x];
  __syncthreads();

  const int wave = threadIdx.x >> 5, lane = threadIdx.x & 31;
  const int half = lane >> 4, lm = lane & 15;
  const int row0 = blockIdx.x * 128 + wave * 16;
  if (row0 >= N) return;
  const int arow = row0 + lm;

  const float* ap = H + (size_t)arow * N_O + half * 2;
  v2f a[8];
#pragma unroll
  for (int t8 = 0; t8 < 8; ++t8) a[t8] = *(const v2f*)(ap + 4 * t8);

  v8f acc = {};
#pragma unroll
  for (int t8 = 0; t8 < 8; ++t8) {
    const int k = 4 * t8 + half * 2;
    v2f b;
    b.x = sBm[(k + 0) * N_F + lm];
    b.y = sBm[(k + 1) * N_F + lm];
    acc = __builtin_amdgcn_wmma_f32_16x16x4_f32(false, a[t8], false, b, (short)0,
                                                acc, false, false);
  }
#pragma unroll
  for (int r = 0; r < 8; ++r) {
    int grow = row0 + r + 8 * half;
    int f = lm;
    size_t p = (size_t)grow * N_F + f;
    float v = acc[r] + sB[f] + outb[p];
    outb[p] = v;
    dout[(size_t)grow * (N_F * N_P) + f * N_P + t] = v;
  }
}

// ------------------------------- host driver --------------------------------

static inline int cdiv_i(int a, int b) { return (a + b - 1) / b; }

extern "C" void kernel_launch(void* const* d_in, const int* in_sizes, int n_in,
                              void* d_out, int out_size, void* d_ws, size_t ws_size,
                              hipStream_t stream) {
  (void)n_in; (void)out_size; (void)ws_size;
  const float* x    = (const float*)d_in[0];
  const int*   eidx = (const int*)d_in[1];
  const float* ew   = (const float*)d_in[2];
  const float* encW = (const float*)d_in[3];
  const float* encb = (const float*)d_in[4];
  const float* decW = (const float*)d_in[5];
  const float* decb = (const float*)d_in[6];
  const float* linW = (const float*)d_in[7];
  const float* linb = (const float*)d_in[8];
  float* dout = (float*)d_out;

  const int E = in_sizes[2];
  const int N = in_sizes[0] / (N_F * N_P);
  const int* srcI = eidx;
  const int* dstI = eidx + E;

  // scratch layout (floats)
  float* ws = (float*)d_ws;
  size_t off = 0;
  float* dego  = ws + off; off += N;           // deg_out
  float* degi  = ws + off; off += N;           // deg_in (adjacent -> single zero)
  float* wno   = ws + off; off += E;
  float* wni   = ws + off; off += E;
  float* XH    = ws + off; off += (size_t)N * N_C;
  float* T1o   = ws + off; off += (size_t)N * N_C;
  float* T1i   = ws + off; off += (size_t)N * N_C;  // adjacent to T1o
  float* Zb    = ws + off; off += (size_t)N * N_O;
  float* Rb    = ws + off; off += (size_t)N * N_O;
  float* Hb    = ws + off; off += (size_t)N * N_O;
  float* outb  = ws + off; off += (size_t)N * N_F;
  float* WzrE  = ws + off; off += KTOT * 64;
  float* WhE   = ws + off; off += KTOT * 32;
  float* WzrD  = ws + off; off += KTOT * 64;
  float* WhD   = ws + off; off += KTOT * 32;
  float* bzrE  = ws + off; off += 64;
  float* bhE   = ws + off; off += 32;
  float* bzrD  = ws + off; off += 64;
  float* bhD   = ws + off; off += 32;

  const int TB = 256;

  // --- graph normalization ---
  zero_kernel<<<cdiv_i(2 * N, TB), TB, 0, stream>>>(dego, 2 * N);
  deg_kernel<<<cdiv_i(E, TB), TB, 0, stream>>>(srcI, dstI, ew, dego, degi, E);
  norm_kernel<<<cdiv_i(E, TB), TB, 0, stream>>>(srcI, dstI, ew, dego, degi, wno, wni, E);

  // --- weight packing ---
  pack_zr_kernel<<<cdiv_i(KTOT * 64, TB), TB, 0, stream>>>(encW, encb, WzrE, bzrE);
  pack_h_kernel <<<cdiv_i(KTOT * 32, TB), TB, 0, stream>>>(encW, encb, WhE, bhE);
  pack_zr_kernel<<<cdiv_i(KTOT * 64, TB), TB, 0, stream>>>(decW, decb, WzrD, bzrD);
  pack_h_kernel <<<cdiv_i(KTOT * 32, TB), TB, 0, stream>>>(decW, decb, WhD, bhD);

  // --- h0 = 0 ---
  zero_kernel<<<cdiv_i(N * N_O, TB), TB, 0, stream>>>(Hb, N * N_O);

  const int gNC   = cdiv_i(N * N_C, TB);
  const int gNO   = cdiv_i(N * N_O, TB);
  const int gZT   = cdiv_i(2 * N * N_C, TB);
  const int gProp = cdiv_i(E * 12, TB);
  const int gZR   = N / 32;
  const int gH    = N / 64;
  const int gLin  = cdiv_i(N, 128);

  // --- encoder: 6 GRU cells with ReLU ---
  for (int t = 0; t < N_P; ++t) {
    build_xh_x_kernel<<<gNC, TB, 0, stream>>>(x, Hb, XH, t, N);
    zero_kernel<<<gZT, TB, 0, stream>>>(T1o, 2 * N * N_C);
    prop_kernel<<<gProp, TB, 0, stream>>>(XH, srcI, dstI, wno, wni, T1o, T1i, E);
    gemm_zr_kernel<<<gZR, TB, 0, stream>>>(XH, T1o, T1i, WzrE, bzrE, Zb, Rb, N);
    update_rh_kernel<<<gNO, TB, 0, stream>>>(Rb, Hb, XH, N);
    zero_kernel<<<gZT, TB, 0, stream>>>(T1o, 2 * N * N_C);
    prop_kernel<<<gProp, TB, 0, stream>>>(XH, srcI, dstI, wno, wni, T1o, T1i, E);
    gemm_h_kernel<<<gH, TB, 0, stream>>>(XH, T1o, T1i, WhE, bhE, Zb, Hb, /*relu=*/1, N);
  }

  // --- decoder: 6 GRU cells + output accumulation ---
  init_out_kernel<<<cdiv_i(N * N_F, TB), TB, 0, stream>>>(x, outb, N);
  for (int t = 0; t < N_P; ++t) {
    build_xh_out_kernel<<<gNC, TB, 0, stream>>>(outb, Hb, XH, N);
    zero_kernel<<<gZT, TB, 0, stream>>>(T1o, 2 * N * N_C);
    prop_kernel<<<gProp, TB, 0, stream>>>(XH, srcI, dstI, wno, wni, T1o, T1i, E);
    gemm_zr_kernel<<<gZR, TB, 0, stream>>>(XH, T1o, T1i, WzrD, bzrD, Zb, Rb, N);
    update_rh_kernel<<<gNO, TB, 0, stream>>>(Rb, Hb, XH, N);
    zero_kernel<<<gZT, TB, 0, stream>>>(T1o, 2 * N * N_C);
    prop_kernel<<<gProp, TB, 0, stream>>>(XH, srcI, dstI, wno, wni, T1o, T1i, E);
    gemm_h_kernel<<<gH, TB, 0, stream>>>(XH, T1o, T1i, WhD, bhD, Zb, Hb, /*relu=*/0, N);
    gemm_lin_kernel<<<gLin, TB, 0, stream>>>(Hb, linW, linb, outb, dout, t, N);
  }
}